// GMM_42691974922667
// MI455X (gfx1250) — compile-verified
//
#include <hip/hip_runtime.h>

typedef float v2f __attribute__((ext_vector_type(2)));
typedef float v8f __attribute__((ext_vector_type(8)));

#define LOG2E_F   1.44269504088896340736f
#define LN2PI_F   1.83787706640934548356f   /* ln(2*pi) */

// One wave handles 16 points per tile.
//   B (P matrix, 4x16 per step): lane l<16 -> p(point l, k=2h), p(point l, k=2h+1)
//   A (coeff matrix, 16x4 per step): lane l -> row (l&15), same k slots
//   D (16x16): lane n (0..15) holds rows 0..5 in acc[0..5] = the six reductions for point n
__global__ __launch_bounds__(256) void gmm_hscore_wmma(
    const float* __restrict__ x,
    const float* __restrict__ mean,
    const float* __restrict__ logvar,
    const float* __restrict__ logweight,
    float* __restrict__ out,
    int n, int tilesPerWave)
{
    const int lane = threadIdx.x & 31;
    const int j    = lane & 15;   // A-row index (reduction id) / point slot
    const int h    = lane >> 4;   // which K-half this lane covers

    // --- one-time per-lane setup: exponent polynomial + A-matrix coefficients ---
    float q2[4], q1[4], q0[4];
    float av[4];
#pragma unroll
    for (int s = 0; s < 2; ++s) {
#pragma unroll
        for (int i = 0; i < 2; ++i) {
            const int k   = 4 * s + 2 * h + i;   // this lane's component for (step s, vgpr i)
            const int idx = 2 * s + i;
            const float mu = mean[k];
            const float lv = logvar[k];
            const float lw = logweight[k];
            const float iv = __expf(-lv);        // 1/var
            // p = exp2(q2*x^2 + q1*x + q0), coefficients pre-scaled by log2(e)
            q2[idx] = -0.5f * iv * LOG2E_F;
            q1[idx] = mu * iv * LOG2E_F;
            q0[idx] = (lw - 0.5f * (lv + LN2PI_F) - 0.5f * iv * mu * mu) * LOG2E_F;
            // coefficient matrix rows: {1, iv, iv*mu, iv^2, mu*iv^2, mu^2*iv^2 - iv, 0...}
            const float iv2 = iv * iv;
            av[idx] = (j == 0) ? 1.0f
                    : (j == 1) ? iv
                    : (j == 2) ? iv * mu
                    : (j == 3) ? iv2
                    : (j == 4) ? mu * iv2
                    : (j == 5) ? (mu * mu * iv2 - iv)
                    : 0.0f;
        }
    }
    v2f a_s0, a_s1;
    a_s0[0] = av[0]; a_s0[1] = av[1];
    a_s1[0] = av[2]; a_s1[1] = av[3];

    const int wave = (int)((blockIdx.x * blockDim.x + threadIdx.x) >> 5);

    // trip count computed once: no per-iteration bounds branch in the loop body
    const int tilesAvail = (n >> 4) - wave * tilesPerWave;
    const int tiles = (tilesAvail < tilesPerWave) ? tilesAvail : tilesPerWave;
    if (tiles <= 0) return;

    int idx = wave * tilesPerWave * 16 + j;     // this lane's x index for tile 0
    float xv = x[idx];                          // pipeline stage 0

    for (int t = 0; t < tiles; ++t) {
        // ---- issue next tile's load early (clamped on last iter); consumed next iter ----
        const int nextIdx = idx + 16;
        const int pfIdx   = (t + 1 < tiles) ? nextIdx : idx;
        const float xnext = x[pfIdx];
        // stream-ahead prefetch (speculative, OOB-safe, no counter traffic)
        __builtin_prefetch(&x[idx + 128], 0, 0);

        const float x2 = xv * xv;

        // probabilities for this lane's 4 (point,k) pairs: 2 FMA + 1 v_exp_f32 each
        v2f b0, b1;
        b0[0] = __builtin_amdgcn_exp2f(fmaf(q1[0], xv, fmaf(q2[0], x2, q0[0])));
        b0[1] = __builtin_amdgcn_exp2f(fmaf(q1[1], xv, fmaf(q2[1], x2, q0[1])));
        b1[0] = __builtin_amdgcn_exp2f(fmaf(q1[2], xv, fmaf(q2[2], x2, q0[2])));
        b1[1] = __builtin_amdgcn_exp2f(fmaf(q1[3], xv, fmaf(q2[3], x2, q0[3])));

        // six K-reductions as [16x8]x[8x16] GEMM: two K=4 WMMA steps
        v8f acc = {};
        acc = __builtin_amdgcn_wmma_f32_16x16x4_f32(false, a_s0, false, b0,
                                                    (short)0, acc, false, false);
        acc = __builtin_amdgcn_wmma_f32_16x16x4_f32(false, a_s1, false, b1,
                                                    (short)0, acc, false, false);

        // lanes 0..15: acc[0..5] = {mpdf, B, A, C, D, E} for point idx
        const float mpdf = acc[0];
        const float Bv   = acc[1];
        const float Av   = acc[2];
        const float Cv   = acc[3];
        const float Dv   = acc[4];
        const float Ev   = acc[5];

        const float mdpdf = fmaf(-Bv, xv, Av);                      // A - x*B
        const float ddpdf = fmaf(-2.0f * xv, Dv, fmaf(Cv, x2, Ev)); // x^2*C - 2x*D + E
        const float rp    = __builtin_amdgcn_rcpf(mpdf);
        const float dln   = mdpdf * rp;
        const float hs    = fmaf(-0.5f * dln, dln, ddpdf * rp);

        if (lane < 16) out[idx] = hs;

        xv  = xnext;        // rotate pipeline
        idx = nextIdx;
    }
}

extern "C" void kernel_launch(void* const* d_in, const int* in_sizes, int n_in,
                              void* d_out, int out_size, void* d_ws, size_t ws_size,
                              hipStream_t stream) {
    const float* x         = (const float*)d_in[0];
    const float* mean      = (const float*)d_in[1];
    const float* logvar    = (const float*)d_in[2];
    const float* logweight = (const float*)d_in[3];
    float* out = (float*)d_out;
    const int n = in_sizes[0];

    const int TPW   = 16;                   // tiles (of 16 points) per wave
    const int block = 256;                  // 8 waves per block
    const int ptsPerBlock = (block / 32) * TPW * 16;  // 2048 points
    const int grid  = (n + ptsPerBlock - 1) / ptsPerBlock;

    gmm_hscore_wmma<<<grid, block, 0, stream>>>(x, mean, logvar, logweight, out, n, TPW);
}